// mLstm_10608569221507
// MI455X (gfx1250) — compile-verified
//
#include <hip/hip_runtime.h>
#include <math.h>

// Sizes from the reference
#define B_   16
#define PLEN 128
#define QLEN 256
#define D_   128
#define H_   128

typedef __attribute__((ext_vector_type(2))) float v2f;
typedef __attribute__((ext_vector_type(8))) float v8f;

// Hardware TRANS tanh on gfx1250 if the builtin exists (name differs across
// clang versions); fall back to libm so compilation never breaks.
#if __has_builtin(__builtin_amdgcn_tanhf)
#define FAST_TANH(x) __builtin_amdgcn_tanhf(x)
#elif __has_builtin(__builtin_amdgcn_tanh_f32)
#define FAST_TANH(x) __builtin_amdgcn_tanh_f32(x)
#else
#define FAST_TANH(x) tanhf(x)
#endif

__device__ __forceinline__ float sigm(float x) { return 1.0f / (1.0f + __expf(-x)); }

// One 16x16 output tile of D += A[16xK] * B[KxN-tile] using V_WMMA_F32_16X16X4_F32.
// A: row-major, leading dim lda (LDS or global); a lane's two A values are
// contiguous (compiler merges to b64).
// Bpk: weight pre-packed into fragment order:
//   Bpk[((k>>1)*N + n)*2 + (k&1)] == W[n][k]
// so a lane's two B values are one aligned 8-byte load.
// Fragment layouts per CDNA5 ISA 7.12.2:
//   A lane L: m = L&15, k = kbase + (L>>4)*2 + v
//   B lane L: k = kbase + (L>>4)*2 + v, n = L&15
__device__ __forceinline__ v8f wmma_k(const float* __restrict__ A, int lda,
                                      const float* __restrict__ Bpk, int N,
                                      int n0, int K, v8f acc, int lane) {
  const int m  = lane & 15;
  const int kh = (lane >> 4) << 1;
  for (int k0 = 0; k0 < K; k0 += 4) {
    v2f a = *(const v2f*)(A + m * lda + k0 + kh);
    v2f b = *(const v2f*)(Bpk + (size_t)(((k0 + kh) >> 1) * N + n0 + m) * 2);
    acc = __builtin_amdgcn_wmma_f32_16x16x4_f32(false, a, false, b, (short)0, acc,
                                                false, false);
  }
  return acc;
}

// Pack W[N][K] (row-major) into WMMA-B fragment order (see wmma_k).
// Output index o = (k>>1)*(2N) + n*2 + (k&1); writes coalesced over o.
__global__ void pack_b_k(const float* __restrict__ in, float* __restrict__ out,
                         int N, int K) {
  int tid = blockIdx.x * blockDim.x + threadIdx.x;
  if (tid < N * K) {
    int kpair = tid / (2 * N);
    int rem   = tid - kpair * 2 * N;
    int n     = rem >> 1;
    int k     = (kpair << 1) | (rem & 1);
    out[tid] = in[n * K + k];
  }
}

// W_hs = q @ Ws^T + Ws_b : [B*Q=4096, 128] GEMM, one wave per 16x16 tile.
__global__ __launch_bounds__(256) void whs_kernel(const float* __restrict__ q,
                                                  const float* __restrict__ WsP,
                                                  const float* __restrict__ Ws_b,
                                                  float* __restrict__ Whs) {
  const int lane  = threadIdx.x & 31;
  const int tile  = blockIdx.x * 8 + (threadIdx.x >> 5);  // 0..2047
  const int mtile = tile >> 3;                            // 0..255 (rows of 16)
  const int ntile = tile & 7;                             // 0..7   (cols of 16)
  v8f acc = {0.f, 0.f, 0.f, 0.f, 0.f, 0.f, 0.f, 0.f};
  acc = wmma_k(q + mtile * 16 * D_, D_, WsP, H_, ntile * 16, D_, acc, lane);
  const int n = ntile * 16 + (lane & 15);
  const float bias = Ws_b[n];
#pragma unroll
  for (int r = 0; r < 8; ++r) {
    int mm = mtile * 16 + r + ((lane >> 4) << 3);
    Whs[mm * H_ + n] = acc[r] + bias;
  }
}

// Persistent recurrent kernel: 1 block, 1024 threads = 32 wave32 waves.
// B=16 batches ride the M=16 dimension of every WMMA.
__global__ __launch_bounds__(1024) void mlstm_kernel(
    const float* __restrict__ p, const float* __restrict__ Whs,
    const float* __restrict__ WtP, const float* __restrict__ WrP,
    const float* __restrict__ WgP, const float* __restrict__ WihP,
    const float* __restrict__ WhhP, const float* __restrict__ Wt_b,
    const float* __restrict__ Wr_b, const float* __restrict__ We_w,
    const float* __restrict__ We_b, const float* __restrict__ Wg_b,
    const float* __restrict__ bih, const float* __restrict__ bhh,
    float* __restrict__ out) {
  // LDS (phase-aliased): 2048 + 4096 + 8192 floats = 56 KB
  __shared__ __align__(16) float h_s[B_ * H_];     // hidden state
  __shared__ __align__(16) float xbuf[B_ * 256];   // [:,0:128]=v then alpha; [:,128:256]=p_t
  __shared__ __align__(16) float region[B_ * 512]; // scores -> gate-lin -> LSTM gates

  const int tid  = threadIdx.x;
  const int lane = tid & 31;
  const int wave = tid >> 5;

  // c lives in registers: thread owns (b,h) = idx and idx+1024 forever.
  float c0 = 0.f, c1 = 0.f;
  for (int i = tid; i < B_ * H_; i += 1024) h_s[i] = 0.f;
  const float web = We_b[0];
  __syncthreads();

  for (int t = 0; t < PLEN; ++t) {
    // stage p_t into xbuf[:,128:256] (it IS the second half of concat x)
    for (int i = tid; i < B_ * D_; i += 1024) {
      int b = i >> 7, d = i & 127;
      xbuf[b * 256 + 128 + d] = p[(b * PLEN + t) * D_ + d];
    }
    __syncthreads();

    // Phase A: v = p_t@Wt^T + h@Wr^T + (Wt_b+Wr_b)  -> xbuf[:,0:128]
    if (wave < 8) {
      int n0 = wave * 16;
      v8f acc = {0.f, 0.f, 0.f, 0.f, 0.f, 0.f, 0.f, 0.f};
      acc = wmma_k(xbuf + 128, 256, WtP, H_, n0, D_, acc, lane);
      acc = wmma_k(h_s, H_, WrP, H_, n0, H_, acc, lane);
      int col = n0 + (lane & 15);
      float bias = Wt_b[col] + Wr_b[col];
#pragma unroll
      for (int r = 0; r < 8; ++r) {
        int mm = r + ((lane >> 4) << 3);
        xbuf[mm * 256 + col] = acc[r] + bias;
      }
    }
    __syncthreads();

    // Phase B: score[b][q] = sum_h tanh(W_hs + v) * We_w  (+ We_b) -> region
#pragma unroll
    for (int r = 0; r < 4; ++r) {
      int row = tid + r * 1024;  // 0..4095 == b*256+q
      int b = row >> 8;
      const float4* w4 = (const float4*)(Whs + row * H_);
      const float4* v4 = (const float4*)(xbuf + b * 256);
      const float4* e4 = (const float4*)We_w;
      float acc = 0.f;
      for (int k = 0; k < H_ / 4; ++k) {
        float4 w = w4[k], vv = v4[k], e = e4[k];
        acc += FAST_TANH(w.x + vv.x) * e.x;
        acc += FAST_TANH(w.y + vv.y) * e.y;
        acc += FAST_TANH(w.z + vv.z) * e.z;
        acc += FAST_TANH(w.w + vv.w) * e.w;
      }
      region[row] = acc + web;
    }
    __syncthreads();

    // Phase C: softmax over q (one wave per batch; 8 elems/lane + shfl_xor tree)
    if (wave < B_) {
      int b = wave;
      float vals[8];
      float m = -1e30f;
#pragma unroll
      for (int j = 0; j < 8; ++j) {
        vals[j] = region[b * 256 + j * 32 + lane];
        m = fmaxf(m, vals[j]);
      }
#pragma unroll
      for (int off = 16; off >= 1; off >>= 1) m = fmaxf(m, __shfl_xor(m, off));
      float s = 0.f;
#pragma unroll
      for (int j = 0; j < 8; ++j) {
        vals[j] = __expf(vals[j] - m);
        s += vals[j];
      }
#pragma unroll
      for (int off = 16; off >= 1; off >>= 1) s += __shfl_xor(s, off);
      float inv = 1.0f / s;
#pragma unroll
      for (int j = 0; j < 8; ++j) region[b * 256 + j * 32 + lane] = vals[j] * inv;
    }
    __syncthreads();

    // Phase D: alpha[b][h] = sum_q a[b][q] * W_hs[b][q][h] -> xbuf[:,0:128]
#pragma unroll
    for (int r = 0; r < 2; ++r) {
      int idx = tid + r * 1024;  // 0..2047
      int b = idx >> 7, hh = idx & 127;
      float acc = 0.f;
      for (int qi = 0; qi < QLEN; ++qi)
        acc += region[b * 256 + qi] * Whs[(b * 256 + qi) * H_ + hh];
      xbuf[b * 256 + hh] = acc;  // x = [alpha, p_t] now complete in xbuf
    }
    __syncthreads();

    // Phase E: glin = x @ Wg^T -> region ; then x *= sigmoid(glin + Wg_b)
    if (wave < 16) {
      int n0 = wave * 16;
      v8f acc = {0.f, 0.f, 0.f, 0.f, 0.f, 0.f, 0.f, 0.f};
      acc = wmma_k(xbuf, 256, WgP, 256, n0, 256, acc, lane);
      int col = n0 + (lane & 15);
#pragma unroll
      for (int r = 0; r < 8; ++r) {
        int mm = r + ((lane >> 4) << 3);
        region[mm * 256 + col] = acc[r];
      }
    }
    __syncthreads();
    for (int i = tid; i < B_ * 256; i += 1024)
      xbuf[i] *= sigm(region[i] + Wg_b[i & 255]);
    __syncthreads();

    // Phase F: gates = x@Wih^T + h@Whh^T + bih + bhh -> region (all 32 waves)
    {
      int n0 = wave * 16;  // 0..511 across 32 waves
      v8f acc = {0.f, 0.f, 0.f, 0.f, 0.f, 0.f, 0.f, 0.f};
      acc = wmma_k(xbuf, 256, WihP, 512, n0, 256, acc, lane);
      acc = wmma_k(h_s, H_, WhhP, 512, n0, H_, acc, lane);
      int col = n0 + (lane & 15);
      float bias = bih[col] + bhh[col];
#pragma unroll
      for (int r = 0; r < 8; ++r) {
        int mm = r + ((lane >> 4) << 3);
        region[mm * 512 + col] = acc[r] + bias;
      }
    }
    __syncthreads();

    // Phase G: LSTM cell update (i,f,g,o order), write h and output
    {
      int idx = tid;
      int b = idx >> 7, hh = idx & 127;
      float gi = region[b * 512 + hh];
      float gf = region[b * 512 + 128 + hh];
      float gg = region[b * 512 + 256 + hh];
      float go = region[b * 512 + 384 + hh];
      float cn = sigm(gf) * c0 + sigm(gi) * FAST_TANH(gg);
      float hn = sigm(go) * FAST_TANH(cn);
      c0 = cn;
      h_s[b * H_ + hh] = hn;
      out[(b * PLEN + t) * H_ + hh] = hn;

      idx = tid + 1024;
      b = idx >> 7;
      hh = idx & 127;
      gi = region[b * 512 + hh];
      gf = region[b * 512 + 128 + hh];
      gg = region[b * 512 + 256 + hh];
      go = region[b * 512 + 384 + hh];
      cn = sigm(gf) * c1 + sigm(gi) * FAST_TANH(gg);
      hn = sigm(go) * FAST_TANH(cn);
      c1 = cn;
      h_s[b * H_ + hh] = hn;
      out[(b * PLEN + t) * H_ + hh] = hn;
    }
    __syncthreads();
  }
}

extern "C" void kernel_launch(void* const* d_in, const int* in_sizes, int n_in,
                              void* d_out, int out_size, void* d_ws, size_t ws_size,
                              hipStream_t stream) {
  (void)in_sizes; (void)n_in; (void)out_size; (void)ws_size;
  const float* p    = (const float*)d_in[0];
  const float* q    = (const float*)d_in[1];
  const float* Ws_w = (const float*)d_in[2];
  const float* Ws_b = (const float*)d_in[3];
  const float* Wt_w = (const float*)d_in[4];
  const float* Wt_b = (const float*)d_in[5];
  const float* Wr_w = (const float*)d_in[6];
  const float* Wr_b = (const float*)d_in[7];
  const float* We_w = (const float*)d_in[8];
  const float* We_b = (const float*)d_in[9];
  const float* Wg_w = (const float*)d_in[10];
  const float* Wg_b = (const float*)d_in[11];
  const float* Wih  = (const float*)d_in[12];
  const float* Whh  = (const float*)d_in[13];
  const float* bih  = (const float*)d_in[14];
  const float* bhh  = (const float*)d_in[15];
  float* out = (float*)d_out;

  // Workspace layout (floats): fragment-packed weights + precomputed W_hs (~3.34 MB)
  float* ws   = (float*)d_ws;
  float* WsP  = ws;                   // 128*128
  float* WtP  = WsP + 128 * 128;      // 128*128
  float* WrP  = WtP + 128 * 128;      // 128*128
  float* WgP  = WrP + 128 * 128;      // 256*256
  float* WihP = WgP + 256 * 256;      // 256*512
  float* WhhP = WihP + 256 * 512;     // 128*512
  float* Whs  = WhhP + 128 * 512;     // 16*256*128

  auto pk = [&](const float* in, float* o, int N, int K) {
    int total = N * K;
    pack_b_k<<<(total + 255) / 256, 256, 0, stream>>>(in, o, N, K);
  };
  pk(Ws_w, WsP, 128, 128);
  pk(Wt_w, WtP, 128, 128);
  pk(Wr_w, WrP, 128, 128);
  pk(Wg_w, WgP, 256, 256);
  pk(Wih, WihP, 512, 256);
  pk(Whh, WhhP, 512, 128);

  // Loop-invariant W_hs = q @ Ws^T + Ws_b (4096x128x128) — grid-parallel WMMA
  whs_kernel<<<256, 256, 0, stream>>>(q, WsP, Ws_b, Whs);

  // Recurrence: single persistent workgroup, 32 waves
  mlstm_kernel<<<1, 1024, 0, stream>>>(p, Whs, WtP, WrP, WgP, WihP, WhhP, Wt_b,
                                       Wr_b, We_w, We_b, Wg_b, bih, bhh, out);
}